// SmallSMBlock3ColorsPY_44753559224599
// MI455X (gfx1250) — compile-verified
//
#include <hip/hip_runtime.h>

typedef __attribute__((ext_vector_type(2))) float v2f;
typedef __attribute__((ext_vector_type(8))) float v8f;

#define TILE 16
#define HALO 18      // TILE + 2 halo
#define HP 512
#define WP 512
#define NB 16        // batch images
#define NIMG 3       // conv input channels
#define NCH 9        // conv output channels (3x3 taps)

// Async global -> LDS copy (CDNA5, ASYNCcnt-tracked). lds_addr is the
// wave-relative LDS byte address (= low 32 bits of a generic LDS pointer).
__device__ __forceinline__ void async_g2l_b32(unsigned int lds_addr,
                                              unsigned long long gaddr) {
    asm volatile("global_load_async_to_lds_b32 %0, %1, off"
                 :: "v"(lds_addr), "v"(gaddr)
                 : "memory");
}

__device__ __forceinline__ void wait_asynccnt0() {
    asm volatile("s_wait_asynccnt 0x0" ::: "memory");
}

__global__ __launch_bounds__(256) void fused_dynconv_kernel(
    const float* __restrict__ image,   // (3,512,512)
    const float* __restrict__ x,       // (16,1,512,512)
    const float* __restrict__ Wg,      // (9,3,3,3) flat: c*27 + i*9 + di*3 + dj
    const float* __restrict__ bg,      // (9,)
    float* __restrict__ y)             // (16,1,512,512)
{
    __shared__ float imgs[NIMG][HALO][HALO];     // zero-padded image tile
    __shared__ float Wlds[16][28];               // padded weight matrix (zeros)
    __shared__ float blds[16];                   // padded bias
    __shared__ float Klds[NCH][TILE][TILE];      // per-pixel dynamic kernels
    __shared__ float xt[NB][HALO][HALO + 1];     // x tiles for all 16 batches

    const int tid = (int)threadIdx.x;
    const int h0 = (int)blockIdx.y * TILE;
    const int w0 = (int)blockIdx.x * TILE;

    // ---- Stage image halo tile + weights + bias into LDS ----
    for (int e = tid; e < NIMG * HALO * HALO; e += 256) {
        int ch = e / (HALO * HALO);
        int r  = (e / HALO) % HALO;
        int c  = e % HALO;
        int gh = h0 + r - 1;
        int gw = w0 + c - 1;
        float v = 0.f;
        if (gh >= 0 && gh < HP && gw >= 0 && gw < WP)
            v = image[(size_t)ch * (HP * WP) + (size_t)gh * WP + gw];
        imgs[ch][r][c] = v;
    }
    for (int e = tid; e < 16 * 28; e += 256) {
        int mm = e / 28;
        int kk = e % 28;
        Wlds[mm][kk] = (mm < NCH && kk < 27) ? Wg[mm * 27 + kk] : 0.f;
    }
    if (tid < 16) blds[tid] = (tid < NCH) ? bg[tid] : 0.f;

    __syncthreads();

    // ---- Kick off async x-tile staging for all 16 batches.
    //      These memory->LDS copies stream in while the WMMA phase runs;
    //      completion is enforced by s_wait_asynccnt + the 2nd barrier. ----
    {
        const unsigned int xt_batch_stride =
            (unsigned int)(sizeof(float) * HALO * (HALO + 1));
        for (int e = tid; e < HALO * HALO; e += 256) {
            const int r = e / HALO;
            const int c = e % HALO;
            const int gh = h0 + r - 1;
            const int gw = w0 + c - 1;
            const bool ib = (gh >= 0) && (gh < HP) && (gw >= 0) && (gw < WP);
            if (ib) {
                unsigned long long gp = (unsigned long long)(uintptr_t)
                    (x + (size_t)gh * WP + (size_t)gw);
                unsigned int la = (unsigned int)(uintptr_t)(&xt[0][r][c]);
#pragma unroll
                for (int n = 0; n < NB; ++n) {
                    async_g2l_b32(la, gp);
                    la += xt_batch_stride;
                    gp += (unsigned long long)(sizeof(float) * HP * WP);
                }
            } else {
#pragma unroll
                for (int n = 0; n < NB; ++n) xt[n][r][c] = 0.f;
            }
        }
    }

    // ---- WMMA im2col GEMM: K[c,p] = sum_k W[c,k] * imgpatch[k,p] ----
    // M = out channel (9 pad 16), N = 16 pixels of one row, K-dim = 27 pad 28.
    const int lane = tid & 31;
    const int wv   = tid >> 5;       // 8 waves
    const int mI   = lane & 15;      // A-row (M) and B-col (N = pixel)
    const int hk   = lane >> 4;      // lane half
    const int rr0  = wv * 2;         // each wave handles 2 pixel rows
    const int rr1  = rr0 + 1;

#if __has_builtin(__builtin_amdgcn_wmma_f32_16x16x4_f32)
    v8f acc0 = {};
    v8f acc1 = {};
#pragma unroll
    for (int s = 0; s < 7; ++s) {
        const int k0  = 4 * s + 2 * hk;          // <= 26, always valid
        const int k1  = k0 + 1;                  // == 27 only for s=6, hk=1 (pad)
        const int k1c = (k1 < 27) ? k1 : 26;     // clamp for safe LDS indexing
        // decode k -> (input channel i, tap di, dj)
        const int i0 = k0 / 9,  q0 = k0 % 9,  di0 = q0 / 3, dj0 = q0 % 3;
        const int i1 = k1c / 9, q1 = k1c % 9, di1 = q1 / 3, dj1 = q1 % 3;
        // A: lane half hk holds K = 2*hk (VGPR0) and K = 2*hk+1 (VGPR1)
        v2f a;
        a.x = Wlds[mI][k0];
        a.y = Wlds[mI][k1];                      // zero at padded k == 27
        // B: VGPR r, lane half -> K = r + 2*hk; N = lane&15
        v2f b0, b1;
        b0.x = imgs[i0][rr0 + di0][mI + dj0];
        b1.x = imgs[i0][rr1 + di0][mI + dj0];
        float t0 = imgs[i1][rr0 + di1][mI + dj1];
        float t1 = imgs[i1][rr1 + di1][mI + dj1];
        b0.y = (k1 < 27) ? t0 : 0.f;
        b1.y = (k1 < 27) ? t1 : 0.f;
        acc0 = __builtin_amdgcn_wmma_f32_16x16x4_f32(false, a, false, b0,
                                                     (short)0, acc0, false, false);
        acc1 = __builtin_amdgcn_wmma_f32_16x16x4_f32(false, a, false, b1,
                                                     (short)0, acc1, false, false);
    }
    // D layout: VGPR r -> channel c = r + 8*lane_half, pixel = lane&15
#pragma unroll
    for (int r = 0; r < 8; ++r) {
        const int c = r + 8 * hk;
        if (c < NCH) {
            Klds[c][rr0][mI] = acc0[r] + blds[c];
            Klds[c][rr1][mI] = acc1[r] + blds[c];
        }
    }
#else
    // Scalar fallback if the f32 WMMA builtin is unavailable
    for (int e = tid; e < NCH * TILE * TILE; e += 256) {
        int c  = e / (TILE * TILE);
        int rr = (e / TILE) % TILE;
        int p  = e % TILE;
        float acc = blds[c];
        for (int kk = 0; kk < 27; ++kk) {
            int i = kk / 9, q = kk % 9, di = q / 3, dj = q % 3;
            acc += Wlds[c][kk] * imgs[i][rr + di][p + dj];
        }
        Klds[c][rr][p] = acc;
    }
#endif

    // All async x-tile copies must have landed before any thread reads xt.
    wait_asynccnt0();
    __syncthreads();

    // ---- Apply per-pixel dynamic kernels to all 16 batch images ----
    const int rr = tid >> 4;   // pixel row in tile
    const int p  = tid & 15;   // pixel col in tile
    float kf[9];
#pragma unroll
    for (int t = 0; t < 9; ++t) kf[t] = Klds[t][rr][p];
    const size_t base = (size_t)(h0 + rr) * WP + (size_t)(w0 + p);
#pragma unroll
    for (int n = 0; n < NB; ++n) {
        float s = 0.f;
#pragma unroll
        for (int di = 0; di < 3; ++di)
#pragma unroll
            for (int dj = 0; dj < 3; ++dj)
                s += xt[n][rr + di][p + dj] * kf[di * 3 + dj];
        y[(size_t)n * (HP * WP) + base] = s;
    }
}

extern "C" void kernel_launch(void* const* d_in, const int* in_sizes, int n_in,
                              void* d_out, int out_size, void* d_ws, size_t ws_size,
                              hipStream_t stream) {
    (void)in_sizes; (void)n_in; (void)out_size; (void)d_ws; (void)ws_size;
    const float* image = (const float*)d_in[0];   // (3,512,512)
    const float* x     = (const float*)d_in[1];   // (16,1,512,512)
    const float* Wg    = (const float*)d_in[2];   // (9,3,3,3)
    const float* bg    = (const float*)d_in[3];   // (9,)
    float* y           = (float*)d_out;           // (16,1,512,512)

    dim3 grid(WP / TILE, HP / TILE);              // 32 x 32 tiles
    fused_dynconv_kernel<<<grid, 256, 0, stream>>>(image, x, Wg, bg, y);
}